// Model_74981539054073
// MI455X (gfx1250) — compile-verified
//
#include <hip/hip_runtime.h>
#include <math.h>

// ---------------- problem constants ----------------
#define N_NODES 2048
#define E_EDGES 65536
#define F_IN    148
#define F_HID   128
#define F_G     148
#define F_G_PAD 160
#define K_ITER  768      // alpha^768 ~ 1e-9 << fp32 eps of accumulated sum
#define SPLITK  8

typedef __attribute__((ext_vector_type(2))) float v2f;
typedef __attribute__((ext_vector_type(8))) float v8f;

// D = A(16x4,f32) * B(4x16,f32) + C(16x16,f32)  -> v_wmma_f32_16x16x4_f32
__device__ __forceinline__ v8f wmma4(v2f a, v2f b, v8f c) {
  return __builtin_amdgcn_wmma_f32_16x16x4_f32(false, a, false, b, (short)0, c,
                                               false, false);
}

// ---- gfx1250 async global->LDS copy (ASYNCcnt-tracked) --------------------
#if __has_builtin(__builtin_amdgcn_global_load_async_to_lds_b128) && \
    __has_builtin(__builtin_amdgcn_s_wait_asynccnt)
#define HAVE_ASYNC_LDS 1
typedef int v4i __attribute__((vector_size(16)));
typedef __attribute__((address_space(1))) v4i* gp_v4i;
typedef __attribute__((address_space(3))) v4i* lp_v4i;
__device__ __forceinline__ void async_copy_b128(void* lds_dst, const float* gsrc) {
  // per-lane: copies 16 bytes from the global address in gsrc to the LDS address in lds_dst, tracked with ASYNCcnt
  __builtin_amdgcn_global_load_async_to_lds_b128(
      (gp_v4i)(const_cast<float*>(gsrc)), (lp_v4i)lds_dst, 0, 0);
}
__device__ __forceinline__ void async_wait0() {
  __builtin_amdgcn_s_wait_asynccnt(0);
}
#else
#define HAVE_ASYNC_LDS 0
#warning "gfx1250 async-to-LDS builtins unavailable; using sync LDS staging fallback"
__device__ __forceinline__ void async_wait0() {}
#endif

// ---------------- workspace layout (floats) ----------------
#define MS_OFF   ((size_t)0)                               // 2048*2048
#define X0_OFF   (MS_OFF   + (size_t)N_NODES * N_NODES)    // 2048*128
#define YA_OFF   (X0_OFF   + (size_t)N_NODES * F_HID)
#define YB_OFF   (YA_OFF   + (size_t)N_NODES * F_HID)
#define XG_OFF   (YB_OFF   + (size_t)N_NODES * F_HID)      // 2048*160
#define DPS_OFF  (XG_OFF   + (size_t)N_NODES * F_G_PAD)    // 2048
#define SC_OFF   (DPS_OFF  + (size_t)N_NODES)              // 8 scalars
#define PART_OFF (SC_OFF   + (size_t)8)                    // SPLITK*2048*128
#define LP_OFF   (PART_OFF + (size_t)SPLITK * N_NODES * F_HID) // 2048
#define POOL_OFF (LP_OFF   + (size_t)2048)                 // 160

// ---------------- utility kernels ----------------
__global__ void k_fill(float* p, size_t n, float v) {
  size_t i = (size_t)blockIdx.x * blockDim.x + threadIdx.x;
  size_t stride = (size_t)gridDim.x * blockDim.x;
  for (; i < n; i += stride) p[i] = v;
}

__global__ void k_scatter(const long long* __restrict__ ei, float* __restrict__ adj) {
  int e = blockIdx.x * blockDim.x + threadIdx.x;
  if (e < E_EDGES) {
    int i = (int)ei[e];
    int j = (int)ei[E_EDGES + e];
    adj[(size_t)i * N_NODES + j] = 1.0f;   // duplicates collapse (set semantics)
    adj[(size_t)j * N_NODES + i] = 1.0f;
  }
}

__global__ void k_dps(const float* __restrict__ adj, float* __restrict__ dps) {
  __shared__ float red[256];
  int row = blockIdx.x;
  float s = 0.f;
  for (int c = threadIdx.x; c < N_NODES; c += 256)
    s += adj[(size_t)row * N_NODES + c];
  red[threadIdx.x] = s;
  __syncthreads();
  for (int off = 128; off > 0; off >>= 1) {
    if ((int)threadIdx.x < off) red[threadIdx.x] += red[threadIdx.x + off];
    __syncthreads();
  }
  if (threadIdx.x == 0) {
    float d = red[0];
    dps[row] = (d > 0.f) ? (1.0f / sqrtf(d)) : 0.0f;
  }
}

__global__ void k_alpha(const float* __restrict__ ap, float* __restrict__ sc) {
  if (threadIdx.x == 0 && blockIdx.x == 0) {
    float a = ap[0];
    float al = 0.5f + 0.5f * tanhf(2.0f * a);
    sc[0] = al;
    sc[1] = 1.0f - al;
  }
}

// Ms = alpha * (dps_i * adj * dps_j + 1e-8 * I), in place over adj
__global__ void k_scale(float* __restrict__ adj, const float* __restrict__ dps,
                        const float* __restrict__ sc) {
  float al = sc[0];
  size_t total = (size_t)N_NODES * N_NODES;
  size_t stride = (size_t)gridDim.x * blockDim.x;
  for (size_t idx = (size_t)blockIdx.x * blockDim.x + threadIdx.x; idx < total;
       idx += stride) {
    size_t i = idx / N_NODES, j = idx - i * N_NODES;
    float v = adj[idx] * dps[i] * dps[j];
    if (i == j) v += 1e-8f;
    adj[idx] = al * v;
  }
}

// ---------------- x0 = nf[2048x148] @ W1^T[148x128] + b1 ----------------
__global__ void k_x0(const float* __restrict__ nf, const float* __restrict__ w1,
                     const float* __restrict__ b1, float* __restrict__ x0) {
  int wave = (blockIdx.x * blockDim.x + threadIdx.x) >> 5;
  int lane = threadIdx.x & 31;
  int tm = wave >> 3, tn = wave & 7;          // 128 x 8 tiles
  int m0 = tm * 16, n0 = tn * 16;
  int lr = lane & 15;
  int koff = (lane >> 4) << 1;                // 0 or 2
  v8f acc = {};
  const float* arow = nf + (size_t)(m0 + lr) * F_IN;
  const float* brow = w1 + (size_t)(n0 + lr) * F_IN;   // B[k][n] = W1[n][k]
  for (int k = 0; k < F_IN; k += 4) {
    v2f a = *(const v2f*)(arow + k + koff);
    v2f b = *(const v2f*)(brow + k + koff);
    acc = wmma4(a, b, acc);
  }
  float bias = b1[n0 + lr];
  int rbase = (lane >> 4) * 8;
  for (int r = 0; r < 8; ++r)
    x0[(size_t)(m0 + rbase + r) * F_HID + n0 + lr] = acc[r] + bias;
}

// ---------------- split-K propagation: part[s] = Ms[:, ks] @ y[ks, :] ------
// Double-buffered B panel staged via async global->LDS (ASYNCcnt), WMMA consume.
__global__ void k_prop(const float* __restrict__ Ms, const float* __restrict__ y,
                       float* __restrict__ part) {
  __shared__ float Bs[2][32 * F_HID];         // 2 x 16 KB panels
  const int CH = 32;                          // K chunk
  const int NCH = (N_NODES / SPLITK) / CH;    // 8 chunks per 256-wide slice
  int rb = blockIdx.x;                        // 0..15 : 128-row block
  int ks = blockIdx.y;                        // 0..SPLITK-1
  int wave = threadIdx.x >> 5;                // 0..7
  int lane = threadIdx.x & 31;
  int lr = lane & 15;
  int koff = (lane >> 4) << 1;
  int m0 = rb * 128 + wave * 16;
  int k0 = ks * (N_NODES / SPLITK);           // 256-wide K slice

  auto issue = [&](int buf, int kb) {
#if HAVE_ASYNC_LDS
    // 256 threads x 4 x b128 = 16 KB panel, no VGPR round-trip
    const char* src = (const char*)(y + (size_t)(k0 + kb) * F_HID) + threadIdx.x * 16;
    char* dst = (char*)&Bs[buf][0] + threadIdx.x * 16;
#pragma unroll
    for (int i = 0; i < 4; ++i)
      async_copy_b128(dst + i * 4096, (const float*)(src + i * 4096));
#else
    const float4* src4 = (const float4*)(y + (size_t)(k0 + kb) * F_HID);
    float4* dst4 = (float4*)&Bs[buf][0];
    for (int t = threadIdx.x; t < CH * F_HID / 4; t += 256) dst4[t] = src4[t];
#endif
  };

  v8f acc[8];
#pragma unroll
  for (int t = 0; t < 8; ++t) acc[t] = (v8f){};

  issue(0, 0);                                // prologue: panel 0 in flight
  for (int c = 0; c < NCH; ++c) {
    int cur = c & 1;
    async_wait0();                            // my panel-loads into Bs[cur] done
    __syncthreads();                          // all waves' loads visible
    if (c + 1 < NCH) issue(cur ^ 1, (c + 1) * CH);   // overlap next panel fetch

    const float* arow = Ms + (size_t)(m0 + lr) * N_NODES + k0 + c * CH;
    const float* B = &Bs[cur][0];
#pragma unroll
    for (int s = 0; s < 8; ++s) {
      v2f a = *(const v2f*)(arow + 4 * s + koff);
      int kk = 4 * s + koff;
#pragma unroll
      for (int t = 0; t < 8; ++t) {
        v2f b;
        b.x = B[kk * F_HID + t * 16 + lr];
        b.y = B[(kk + 1) * F_HID + t * 16 + lr];
        acc[t] = wmma4(a, b, acc[t]);
      }
    }
    __syncthreads();                          // panel may be re-issued next iter
  }

  int rbase = (lane >> 4) * 8;
  float* pout = part + (size_t)ks * N_NODES * F_HID;
  for (int t = 0; t < 8; ++t)
    for (int r = 0; r < 8; ++r)
      pout[(size_t)(m0 + rbase + r) * F_HID + t * 16 + lr] = acc[t][r];
}

// y_out = x0 + sum_s part[s]   (fixed order -> deterministic)
__global__ void k_red(const float* __restrict__ part, const float* __restrict__ x0,
                      float* __restrict__ yout) {
  size_t total = (size_t)N_NODES * F_HID;
  size_t stride = (size_t)gridDim.x * blockDim.x;
  for (size_t i = (size_t)blockIdx.x * blockDim.x + threadIdx.x; i < total;
       i += stride) {
    float s = x0[i];
#pragma unroll
    for (int k = 0; k < SPLITK; ++k) s += part[(size_t)k * total + i];
    yout[i] = s;
  }
}

// ---------------- xg = y[2048x128] @ Wg^T[128x148] + bg  (padded to 160) ---
__global__ void k_xg(const float* __restrict__ y, const float* __restrict__ wg,
                     const float* __restrict__ bg, float* __restrict__ xg) {
  int wave = (blockIdx.x * blockDim.x + threadIdx.x) >> 5;
  int lane = threadIdx.x & 31;
  const int TN = F_G_PAD / 16;                // 10 col tiles
  int tm = wave / TN, tn = wave % TN;
  int m0 = tm * 16, n0 = tn * 16;
  int lr = lane & 15;
  int koff = (lane >> 4) << 1;
  int col = n0 + lr;
  bool valid = col < F_G;
  v8f acc = {};
  const float* arow = y + (size_t)(m0 + lr) * F_HID;
  const float* brow = wg + (size_t)(valid ? col : 0) * F_HID;
  for (int k = 0; k < F_HID; k += 4) {
    v2f a = *(const v2f*)(arow + k + koff);
    v2f b = *(const v2f*)(brow + k + koff);
    if (!valid) { b.x = 0.f; b.y = 0.f; }     // v_cndmask, EXEC stays full
    acc = wmma4(a, b, acc);
  }
  float bias = valid ? bg[col] : 0.f;
  int rbase = (lane >> 4) * 8;
  for (int r = 0; r < 8; ++r) {
    float v = valid ? (acc[r] + bias) : 0.f;
    xg[(size_t)(m0 + rbase + r) * F_G_PAD + col] = v;
  }
}

// ---------------- loss: sum relu(xg @ xg^T / sqrt(128)) off-diagonal -------
__global__ void k_sim(const float* __restrict__ xg, float* __restrict__ lpart) {
  __shared__ float red[256];
  int wave = threadIdx.x >> 5;
  int gw = blockIdx.x * 8 + wave;             // 16384 tile-waves
  int lane = threadIdx.x & 31;
  int tI = gw >> 7, tJ = gw & 127;
  int m0 = tI * 16, n0 = tJ * 16;
  int lr = lane & 15;
  int koff = (lane >> 4) << 1;
  v8f acc = {};
  const float* arow = xg + (size_t)(m0 + lr) * F_G_PAD;
  const float* brow = xg + (size_t)(n0 + lr) * F_G_PAD;   // B[k][n] = xg[n][k]
  for (int k = 0; k < F_G; k += 4) {
    v2f a = *(const v2f*)(arow + k + koff);
    v2f b = *(const v2f*)(brow + k + koff);
    acc = wmma4(a, b, acc);
  }
  const float scale = 0.08838834764831845f;   // 1/sqrt(128)
  int rbase = (lane >> 4) * 8;
  float s = 0.f;
  for (int r = 0; r < 8; ++r) {
    int gr = m0 + rbase + r, gc = n0 + lr;
    float v = acc[r] * scale;
    v = (gr == gc) ? 0.f : v;
    s += fmaxf(v, 0.f);
  }
  red[threadIdx.x] = s;
  __syncthreads();
  for (int off = 128; off > 0; off >>= 1) {
    if ((int)threadIdx.x < off) red[threadIdx.x] += red[threadIdx.x + off];
    __syncthreads();
  }
  if (threadIdx.x == 0) lpart[blockIdx.x] = red[0];
}

__global__ void k_lossred(const float* __restrict__ lpart, float* __restrict__ out2) {
  __shared__ float red[256];
  float s = 0.f;
  for (int i = threadIdx.x; i < 2048; i += 256) s += lpart[i];
  red[threadIdx.x] = s;
  __syncthreads();
  for (int off = 128; off > 0; off >>= 1) {
    if ((int)threadIdx.x < off) red[threadIdx.x] += red[threadIdx.x + off];
    __syncthreads();
  }
  if (threadIdx.x == 0)
    out2[0] = red[0] / ((float)N_NODES * (float)(N_NODES - 1));
}

// ---------------- pooled = max_m relu(xg * (1-alpha)) ----------------------
__global__ void k_pool(const float* __restrict__ xg, const float* __restrict__ sc,
                       float* __restrict__ pooled) {
  int c = blockIdx.x * blockDim.x + threadIdx.x;
  if (c >= F_G) return;
  float mx = -INFINITY;
  for (int m = 0; m < N_NODES; ++m)
    mx = fmaxf(mx, xg[(size_t)m * F_G_PAD + c]);
  pooled[c] = fmaxf(mx * sc[1], 0.f);         // (1-alpha) > 0
}

__global__ void k_out(const float* __restrict__ pooled, const float* __restrict__ w2,
                      const float* __restrict__ b2, float* __restrict__ out) {
  int j = threadIdx.x;
  if (j < 2) {
    float s = b2[j];
    for (int c = 0; c < F_G; ++c) s += pooled[c] * w2[j * F_G + c];
    out[j] = s;
  }
}

// ---------------- host side ------------------------------------------------
extern "C" void kernel_launch(void* const* d_in, const int* in_sizes, int n_in,
                              void* d_out, int out_size, void* d_ws, size_t ws_size,
                              hipStream_t stream) {
  (void)in_sizes; (void)n_in; (void)out_size; (void)ws_size;
  const float* nf     = (const float*)d_in[0];
  const long long* ei = (const long long*)d_in[1];
  const float* w1     = (const float*)d_in[2];
  const float* b1     = (const float*)d_in[3];
  const float* ap     = (const float*)d_in[4];
  const float* wg     = (const float*)d_in[5];
  const float* bg     = (const float*)d_in[6];
  const float* w2     = (const float*)d_in[7];
  const float* b2     = (const float*)d_in[8];
  float* ws  = (float*)d_ws;
  float* out = (float*)d_out;

  float* adj    = ws + MS_OFF;
  float* x0     = ws + X0_OFF;
  float* yA     = ws + YA_OFF;
  float* yB     = ws + YB_OFF;
  float* xg     = ws + XG_OFF;
  float* dps    = ws + DPS_OFF;
  float* sc     = ws + SC_OFF;
  float* part   = ws + PART_OFF;
  float* lpart  = ws + LP_OFF;
  float* pooled = ws + POOL_OFF;

  // adjacency -> alpha * normalized matrix (in place)
  k_fill<<<4096, 256, 0, stream>>>(adj, (size_t)N_NODES * N_NODES, 0.f);
  k_scatter<<<E_EDGES / 256, 256, 0, stream>>>(ei, adj);
  k_dps<<<N_NODES, 256, 0, stream>>>(adj, dps);
  k_alpha<<<1, 32, 0, stream>>>(ap, sc);
  k_scale<<<2048, 256, 0, stream>>>(adj, dps, sc);

  // x0 = nf @ W1^T + b1
  k_x0<<<128, 256, 0, stream>>>(nf, w1, b1, x0);

  // y = sum_{k=0}^{N} (alpha*M)^k x  via Horner: y <- x0 + Ms @ y
  const float* yin = x0;
  float* cur = yA;
  float* nxt = yB;
  for (int it = 0; it < K_ITER; ++it) {
    k_prop<<<dim3(16, SPLITK), 256, 0, stream>>>(adj, yin, part);
    k_red<<<512, 256, 0, stream>>>(part, x0, cur);
    yin = cur;
    float* t = cur; cur = nxt; nxt = t;
  }

  // xg = y @ Wg^T + bg
  k_xg<<<160, 256, 0, stream>>>(yin, wg, bg, xg);

  // loss
  k_sim<<<2048, 256, 0, stream>>>(xg, lpart);
  k_lossred<<<1, 256, 0, stream>>>(lpart, out + 2);

  // head
  k_pool<<<1, F_G_PAD, 0, stream>>>(xg, sc, pooled);
  k_out<<<1, 32, 0, stream>>>(pooled, w2, b2, out);
}